// GCNLayers_38439957299973
// MI455X (gfx1250) — compile-verified
//
#include <hip/hip_runtime.h>
#include <hip/hip_bf16.h>
#include <cstdint>

#define HID 256

typedef __attribute__((ext_vector_type(2))) float v2f;
typedef __attribute__((ext_vector_type(8))) float v8f;

// ---------------------------------------------------------------------------
// GEMM: Y[N,256] = H[N,256] @ W^T  (W is [256,256], row j = output channel j)
// One wave computes a 16x64 strip of Y with 4 independent f32 WMMA
// accumulators: A tile loaded once per k-step, 4 B tiles, 4 back-to-back
// v_wmma_f32_16x16x4_f32 with disjoint accumulators (RAW chain length 4).
// ---------------------------------------------------------------------------
__global__ __launch_bounds__(256) void gcn_gemm_wmma_f32(
    const float* __restrict__ H, const float* __restrict__ W,
    float* __restrict__ Y, int N, int nRowTiles)
{
    const int wave = blockIdx.x * (blockDim.x >> 5) + (threadIdx.x >> 5);
    const int lane = threadIdx.x & 31;
    const int rt = wave >> 2;          // row tile (16 rows of H)
    const int cg = wave & 3;           // column group (64 cols) : HID/64 == 4
    if (rt >= nRowTiles) return;       // wave-uniform: EXEC stays all-ones

    const int half = lane & 15;
    const int koff = (lane >> 4) << 1; // lanes 0-15 -> K,K+1 ; lanes 16-31 -> K+2,K+3
    int rowA = rt * 16 + half;
    if (rowA >= N) rowA = N - 1;       // clamp loads (partial-tile stores guarded)

    const float* hp = H + (size_t)rowA * HID + koff;
    const float* wp = W + (size_t)(cg * 64 + half) * HID + koff;

    v8f acc0 = {}, acc1 = {}, acc2 = {}, acc3 = {};
#pragma unroll 4
    for (int k = 0; k < HID; k += 4) {
        v2f a  = *(const v2f*)(hp + k);
        v2f b0 = *(const v2f*)(wp + k);
        v2f b1 = *(const v2f*)(wp + 16 * HID + k);
        v2f b2 = *(const v2f*)(wp + 32 * HID + k);
        v2f b3 = *(const v2f*)(wp + 48 * HID + k);
        acc0 = __builtin_amdgcn_wmma_f32_16x16x4_f32(false, a, false, b0, (short)0, acc0, false, false);
        acc1 = __builtin_amdgcn_wmma_f32_16x16x4_f32(false, a, false, b1, (short)0, acc1, false, false);
        acc2 = __builtin_amdgcn_wmma_f32_16x16x4_f32(false, a, false, b2, (short)0, acc2, false, false);
        acc3 = __builtin_amdgcn_wmma_f32_16x16x4_f32(false, a, false, b3, (short)0, acc3, false, false);
    }

    // C/D layout: lanes 0-15 hold M=v, lanes 16-31 hold M=v+8 ; N = lane&15
    const int m0 = (lane >> 4) << 3;
    float* yp = Y + (size_t)(rt * 16 + m0) * HID + cg * 64 + half;

    if (rt * 16 + 16 <= N) {           // full tile: unconditional stores
#pragma unroll
        for (int v = 0; v < 8; ++v) {
            size_t r = (size_t)v * HID;
            yp[r]      = acc0[v];
            yp[r + 16] = acc1[v];
            yp[r + 32] = acc2[v];
            yp[r + 48] = acc3[v];
        }
    } else {                           // partial last tile (not hit when N%16==0)
#pragma unroll
        for (int v = 0; v < 8; ++v) {
            if (rt * 16 + m0 + v < N) {
                size_t r = (size_t)v * HID;
                yp[r]      = acc0[v];
                yp[r + 16] = acc1[v];
                yp[r + 32] = acc2[v];
                yp[r + 48] = acc3[v];
            }
        }
    }
}

// ---------------------------------------------------------------------------
// Degree / normalization (computed once; degree includes the self loop)
// ---------------------------------------------------------------------------
__global__ void deg_init(float* __restrict__ deg, int N)
{
    int i = blockIdx.x * blockDim.x + threadIdx.x;
    if (i < N) deg[i] = 1.0f;   // self loop
}

__global__ void deg_accum(const long long* __restrict__ ei, int E,
                          float* __restrict__ deg, int N)
{
    int e = blockIdx.x * blockDim.x + threadIdx.x;
    if (e < E) {
        int d = (int)ei[(size_t)E + e];
        if (d >= 0 && d < N) unsafeAtomicAdd(&deg[d], 1.0f);
    }
}

__global__ void deg_to_dinv(float* __restrict__ dinv, int N)
{
    int i = blockIdx.x * blockDim.x + threadIdx.x;
    if (i < N) {
        float d = dinv[i];
        dinv[i] = d > 0.0f ? rsqrtf(d) : 0.0f;
    }
}

// ---------------------------------------------------------------------------
// Aggregation: agg[i] = dinv[i]^2 * y[i]  (self loop, also initializes agg)
// then       : agg[dst] += dinv[src]*dinv[dst] * y[src]   per edge (atomics)
// ---------------------------------------------------------------------------
__global__ void agg_self(const float* __restrict__ Y,
                         const float* __restrict__ dinv,
                         float* __restrict__ agg, int N)
{
    size_t t = (size_t)blockIdx.x * blockDim.x + threadIdx.x;
    size_t total = (size_t)N * (HID / 4);
    if (t >= total) return;
    int node = (int)(t >> 6);          // HID/4 == 64 groups per node
    int c    = (int)(t & 63) << 2;
    float di = dinv[node];
    float s  = di * di;
    const float4 y = *(const float4*)(Y + (size_t)node * HID + c);
    float4 o; o.x = y.x * s; o.y = y.y * s; o.z = y.z * s; o.w = y.w * s;
    *(float4*)(agg + (size_t)node * HID + c) = o;
}

__global__ void agg_edges(const float* __restrict__ Y,
                          const long long* __restrict__ ei,
                          const float* __restrict__ dinv,
                          float* __restrict__ agg, int E, int N)
{
    size_t t = (size_t)blockIdx.x * blockDim.x + threadIdx.x;
    size_t total = (size_t)E * 64;
    if (t >= total) return;
    int e = (int)(t >> 6);
    int c = (int)(t & 63) << 2;
    int s = (int)ei[e];
    int d = (int)ei[(size_t)E + e];
    if ((unsigned)s >= (unsigned)N || (unsigned)d >= (unsigned)N) return;
    float w = dinv[s] * dinv[d];
    const float4 y = *(const float4*)(Y + (size_t)s * HID + c);
    float* ap = agg + (size_t)d * HID + c;
    unsafeAtomicAdd(ap + 0, y.x * w);
    unsafeAtomicAdd(ap + 1, y.y * w);
    unsafeAtomicAdd(ap + 2, y.z * w);
    unsafeAtomicAdd(ap + 3, y.w * w);
}

// ---------------------------------------------------------------------------
// Finalize: hout = relu(agg + b) [+ hres]   (safe in-place: hout may == hres)
// ---------------------------------------------------------------------------
__global__ void finalize(const float* __restrict__ agg,
                         const float* __restrict__ b,
                         const float* __restrict__ hres,
                         float* __restrict__ hout,
                         int N, int residual)
{
    size_t idx = (size_t)blockIdx.x * blockDim.x + threadIdx.x;
    size_t total = (size_t)N * HID;
    if (idx >= total) return;
    int c = (int)(idx & (HID - 1));
    float v = agg[idx] + b[c];
    v = fmaxf(v, 0.0f);
    if (residual) v += hres[idx];
    hout[idx] = v;
}

// ---------------------------------------------------------------------------
extern "C" void kernel_launch(void* const* d_in, const int* in_sizes, int n_in,
                              void* d_out, int out_size, void* d_ws, size_t ws_size,
                              hipStream_t stream)
{
    const float*     x  = (const float*)d_in[0];
    const long long* ei = (const long long*)d_in[1];   // int64 per reference
    const float*     W0 = (const float*)d_in[2];
    const float*     b0 = (const float*)d_in[3];
    const float*     W1 = (const float*)d_in[4];
    const float*     b1 = (const float*)d_in[5];
    const float*     W2 = (const float*)d_in[6];
    const float*     b2 = (const float*)d_in[7];
    float* out = (float*)d_out;

    const int N = in_sizes[0] / HID;   // 50000
    const int E = in_sizes[1] / 2;     // 800000
    const size_t nh = (size_t)N * HID;

    // Workspace layout: [h buffer | agg buffer | dinv].  d_out doubles as the
    // per-layer GEMM output (y); it is dead by the time finalize writes it.
    float* Hbuf = (float*)d_ws;        // persistent hidden state
    float* Agg  = Hbuf + nh;           // aggregation accumulator
    float* Dinv = Agg + nh;            // N floats (degree, then rsqrt(degree))
    float* Ybuf = out;                 // GEMM output, reused every layer

    const int TPB = 256;
    const int rowTiles = (N + 15) / 16;
    const int gemmWaves = rowTiles * (HID / 64);     // 16x64 strip per wave
    const int gemmBlocks = (gemmWaves * 32 + TPB - 1) / TPB;
    const int nBlocksN    = (N + TPB - 1) / TPB;
    const int nBlocksE    = (E + TPB - 1) / TPB;
    const long long selfT = (long long)N * 64;
    const int nBlocksSelf = (int)((selfT + TPB - 1) / TPB);
    const long long edgeT = (long long)E * 64;
    const int nBlocksEdge = (int)((edgeT + TPB - 1) / TPB);
    const long long finT  = (long long)N * HID;
    const int nBlocksFin  = (int)((finT + TPB - 1) / TPB);

    // --- normalization (once; degree is layer-invariant) ---
    deg_init   <<<nBlocksN, TPB, 0, stream>>>(Dinv, N);
    deg_accum  <<<nBlocksE, TPB, 0, stream>>>(ei, E, Dinv, N);
    deg_to_dinv<<<nBlocksN, TPB, 0, stream>>>(Dinv, N);

    // --- layer 0: h1 = relu(conv(x)) ---
    gcn_gemm_wmma_f32<<<gemmBlocks, TPB, 0, stream>>>(x, W0, Ybuf, N, rowTiles);
    agg_self  <<<nBlocksSelf, TPB, 0, stream>>>(Ybuf, Dinv, Agg, N);
    agg_edges <<<nBlocksEdge, TPB, 0, stream>>>(Ybuf, ei, Dinv, Agg, E, N);
    finalize  <<<nBlocksFin,  TPB, 0, stream>>>(Agg, b0, Hbuf, Hbuf, N, 0);

    // --- layer 1: h2 = h1 + relu(conv(h1))  (in-place residual update) ---
    gcn_gemm_wmma_f32<<<gemmBlocks, TPB, 0, stream>>>(Hbuf, W1, Ybuf, N, rowTiles);
    agg_self  <<<nBlocksSelf, TPB, 0, stream>>>(Ybuf, Dinv, Agg, N);
    agg_edges <<<nBlocksEdge, TPB, 0, stream>>>(Ybuf, ei, Dinv, Agg, E, N);
    finalize  <<<nBlocksFin,  TPB, 0, stream>>>(Agg, b1, Hbuf, Hbuf, N, 1);

    // --- layer 2: out = h2 + relu(conv(h2)) ---
    gcn_gemm_wmma_f32<<<gemmBlocks, TPB, 0, stream>>>(Hbuf, W2, Ybuf, N, rowTiles);
    agg_self  <<<nBlocksSelf, TPB, 0, stream>>>(Ybuf, Dinv, Agg, N);
    agg_edges <<<nBlocksEdge, TPB, 0, stream>>>(Ybuf, ei, Dinv, Agg, E, N);
    finalize  <<<nBlocksFin,  TPB, 0, stream>>>(Agg, b2, Hbuf, out, N, 1);
}